// SPCoGCNet_25692494365012
// MI455X (gfx1250) — compile-verified
//
#include <hip/hip_runtime.h>
#include <hip/hip_bf16.h>

// ---------------------------------------------------------------------------
// SPCoGCNet forward on gfx1250 (MI455X).
// Matrix math: V_WMMA_F32_16X16X4_F32 (exact fp32, wave32). Each wave owns a
// 16x64 strip of C (4 accumulator tiles) so one A b64 load feeds 4 WMMAs.
// Weights are pre-transposed to [N,K] so B operands are contiguous b64 loads.
// Graph aggregation uses f32 global atomics resolving in the 192MB L2.
// ---------------------------------------------------------------------------

typedef float v2f __attribute__((ext_vector_type(2)));
typedef float v8f __attribute__((ext_vector_type(8)));

#define HH  64    // latent width H
#define HID 128   // MLP hidden width (and encoder/decoder hidden)

// ---------------------------------------------------------------- utilities
__global__ void k_zero(float* __restrict__ p, long n) {
  long t = (long)blockIdx.x * blockDim.x + threadIdx.x;
  if (t < n) p[t] = 0.0f;
}

// Wt[n, k] = W[k, n]
__global__ void k_transpose(const float* __restrict__ W, float* __restrict__ Wt,
                            int K, int N) {
  int t = blockIdx.x * blockDim.x + threadIdx.x;
  if (t >= K * N) return;
  int k = t / N, n = t - k * N;
  Wt[n * K + k] = W[t];
}

// H[m, j] = b1[j] + sum_k X[m, k] * W1[k, j]   (Kin in {1,3}, 128 cols)
__global__ void k_lin_small(const float* __restrict__ X,
                            const float* __restrict__ W1,
                            const float* __restrict__ b1,
                            float* __restrict__ H, int M, int Kin) {
  long t = (long)blockIdx.x * blockDim.x + threadIdx.x;
  if (t >= (long)M * HID) return;
  int m = (int)(t >> 7);
  int j = (int)(t & (HID - 1));
  float acc = b1[j];
  for (int k = 0; k < Kin; ++k)
    acc += X[(long)m * Kin + k] * W1[k * HID + j];
  H[t] = acc;
}

// column sums / sum-of-squares of H [M, 128] -> stats[0..127]=sum, [128..255]=sumsq
__global__ void k_bn_stats(const float* __restrict__ H,
                           float* __restrict__ stats, int M) {
  int col = threadIdx.x;            // blockDim.x == 128
  float s = 0.0f, ss = 0.0f;
  for (int r = blockIdx.x; r < M; r += gridDim.x) {
    float h = H[(long)r * HID + col];
    s += h;
    ss += h * h;
  }
  atomicAdd(&stats[col], s);
  atomicAdd(&stats[HID + col], ss);
}

// in-place: H = relu((H - mean) * rsqrt(var + eps) * g + be)   (biased var)
__global__ void k_bn_norm_relu(float* __restrict__ H,
                               const float* __restrict__ stats,
                               const float* __restrict__ g,
                               const float* __restrict__ be, int M) {
  long t = (long)blockIdx.x * blockDim.x + threadIdx.x;
  if (t >= (long)M * HID) return;
  int col = (int)(t & (HID - 1));
  float invM  = 1.0f / (float)M;
  float mean  = stats[col] * invM;
  float var   = stats[HID + col] * invM - mean * mean;
  float h = H[t];
  h = (h - mean) * rsqrtf(var + 1e-5f) * g[col] + be[col];
  H[t] = fmaxf(h, 0.0f);
}

// msg = relu(v[row] + e) + eps ; agg[col] += msg    (one thread per (edge, c))
__global__ void k_scatter_node(const float* __restrict__ v,
                               const float* __restrict__ e,
                               const int* __restrict__ row,
                               const int* __restrict__ col,
                               float* __restrict__ agg, int E) {
  long t = (long)blockIdx.x * blockDim.x + threadIdx.x;
  if (t >= (long)E * HH) return;
  int i = (int)(t >> 6);
  int c = (int)(t & (HH - 1));
  float m = v[(long)row[i] * HH + c] + e[t];
  m = fmaxf(m, 0.0f) + 1e-7f;
  atomicAdd(&agg[(long)col[i] * HH + c], m);
}

// emsg = relu(e[er]) + eps ; eagg[ec] += emsg   (line graph, ~1.9M edges)
__global__ void k_scatter_edge(const float* __restrict__ e,
                               const int* __restrict__ er,
                               const int* __restrict__ ec,
                               float* __restrict__ eagg, long EL) {
  long t = (long)blockIdx.x * blockDim.x + threadIdx.x;
  if (t >= EL * HH) return;
  long i = t >> 6;
  int  c = (int)(t & (HH - 1));
  float m = fmaxf(e[(long)er[i] * HH + c], 0.0f) + 1e-7f;
  atomicAdd(&eagg[(long)ec[i] * HH + c], m);
}

// out[m, j] = b2[j] + sum_k H[m, k] * W2[k, j],  j in {0,1}
__global__ void k_dec_out(const float* __restrict__ H,
                          const float* __restrict__ W2,
                          const float* __restrict__ b2,
                          float* __restrict__ out, int M) {
  long t = (long)blockIdx.x * blockDim.x + threadIdx.x;
  if (t >= (long)M * 2) return;
  int m = (int)(t >> 1);
  int j = (int)(t & 1);
  float acc = b2[j];
  const float* h = H + (long)m * HID;
  #pragma unroll 8
  for (int k = 0; k < HID; ++k) acc += h[k] * W2[k * 2 + j];
  out[t] = acc;
}

// ---------------------------------------------------------------- WMMA GEMM
// C[M, N] = maybe_relu( (A (+A2))[M, K] @ W[K, N] + bias[N] )
// W is supplied TRANSPOSED: Wt[N, K]  ->  B operand is a contiguous b64 load.
// One wave per 16x64 strip (4 WMMA tiles); K-loop over V_WMMA_F32_16X16X4_F32.
// f32 operand layouts (ISA 7.12.2):
//   A 16x4:  lane 0-15 -> M=lane; lane 16-31 -> M=lane-16; v0=K(2*kh), v1=K(2*kh+1)
//   B 4x16:  lane 0-15 -> N=lane (low K pair), lanes 16-31 -> high K pair
//   C 16x16: VGPR r holds M = r + 8*(lane>>4), N = lane&15
template <int K, bool HASA2>
__global__ void k_gemm_wmma(const float* __restrict__ A,
                            const float* __restrict__ A2,
                            const float* __restrict__ Wt,
                            const float* __restrict__ bias,
                            float* __restrict__ C,
                            int M, int N, int relu) {
  int gtid = blockIdx.x * blockDim.x + threadIdx.x;
  int wave = gtid >> 5;
  int lane = threadIdx.x & 31;
  int wN = N >> 6;                 // 16x64 strips across N
  int tm = wave / wN;              // uniform across the wave
  int tw = wave - tm * wN;
  if (tm * 16 >= M) return;        // wave-uniform exit (padding waves)

  int l15 = lane & 15;
  int kh  = lane >> 4;             // which K half-pair this lane holds
  int rowA = tm * 16 + l15;
  int rA   = rowA < M ? rowA : (M - 1);   // clamp: loads stay in-bounds,
                                          // EXEC stays all-1s for WMMA
  const float* Arow  = A + (long)rA * K;
  const float* A2row = HASA2 ? (A2 + (long)rA * K) : nullptr;
  int nj0 = tw * 64 + l15;
  const float* Wt0 = Wt + (long)nj0 * K;

  v8f acc[4] = {};
  #pragma unroll 4
  for (int k0 = 0; k0 < K; k0 += 4) {
    int ka = k0 + 2 * kh;
    v2f a = *(const v2f*)(Arow + ka);               // 8B-aligned (K,ka even)
    if (HASA2) {
      v2f a2 = *(const v2f*)(A2row + ka);
      a.x += a2.x; a.y += a2.y;
    }
    v2f b0 = *(const v2f*)(Wt0 + ka);
    v2f b1 = *(const v2f*)(Wt0 + 16 * K + ka);
    v2f b2 = *(const v2f*)(Wt0 + 32 * K + ka);
    v2f b3 = *(const v2f*)(Wt0 + 48 * K + ka);
    acc[0] = __builtin_amdgcn_wmma_f32_16x16x4_f32(false, a, false, b0, (short)0, acc[0], false, false);
    acc[1] = __builtin_amdgcn_wmma_f32_16x16x4_f32(false, a, false, b1, (short)0, acc[1], false, false);
    acc[2] = __builtin_amdgcn_wmma_f32_16x16x4_f32(false, a, false, b2, (short)0, acc[2], false, false);
    acc[3] = __builtin_amdgcn_wmma_f32_16x16x4_f32(false, a, false, b3, (short)0, acc[3], false, false);
  }

  #pragma unroll
  for (int t = 0; t < 4; ++t) {
    int nj = nj0 + 16 * t;
    float bj = bias[nj];
    #pragma unroll
    for (int r = 0; r < 8; ++r) {
      int mr = tm * 16 + r + 8 * kh;
      if (mr < M) {
        float val = acc[t][r] + bj;
        if (relu) val = fmaxf(val, 0.0f);
        C[(long)mr * N + nj] = val;
      }
    }
  }
}

// ---------------------------------------------------------------- launcher
static inline int nblk(long n) { return (int)((n + 255) / 256); }

extern "C" void kernel_launch(void* const* d_in, const int* in_sizes, int n_in,
                              void* d_out, int out_size, void* d_ws, size_t ws_size,
                              hipStream_t stream) {
  const int BLK = 256;

  const float* x   = (const float*)d_in[0];
  const int*   ei  = (const int*)  d_in[1];
  const float* ea  = (const float*)d_in[2];
  const int*   eei = (const int*)  d_in[3];

  const int  N  = in_sizes[0] / 3;       // 30000
  const int  E  = in_sizes[2];           // edge_attr is [E, 1]
  const long EL = (long)in_sizes[3] / 2; // line-graph edge count

  const int* row = ei;
  const int* col = ei + E;
  const int* er  = eei;
  const int* ec  = eei + EL;

  // params flattened as a jax pytree (dict keys sorted):
  //   top: edge_dec, edge_enc, layers[0..2]{edge,node}, node_dec, node_enc
  //   mlp: W1, W2, b1, b2, be, g
  auto P = [&](int i) { return (const float*)d_in[i]; };
  const int ED = 4;            // edge_dec base
  const int EE = 4 + 6;        // edge_enc base
  const int ND = 4 + 48;       // node_dec base
  const int NE = 4 + 54;       // node_enc base
  auto LE = [](int l) { return 4 + 12 + 12 * l; };      // layer l edge-mlp base
  auto LN = [](int l) { return 4 + 12 + 12 * l + 6; };  // layer l node-mlp base
  // within mlp base b: W1=b, W2=b+1, b1=b+2, b2=b+3, be=b+4, g=b+5

  // ---- workspace partition (floats) ----
  float* ws = (float*)d_ws;
  long maxME = (long)(N > E ? N : E);
  float* v0    = ws;  ws += (long)N * HH;
  float* v1    = ws;  ws += (long)N * HH;
  float* e0    = ws;  ws += (long)E * HH;
  float* e1    = ws;  ws += (long)E * HH;
  float* agg   = ws;  ws += maxME * HH;
  float* hbuf  = ws;  ws += maxME * HID;
  float* stats = ws;  ws += 2 * HID;
  float* wtbuf = ws;  ws += (long)HID * HH;   // transposed-weight scratch
  (void)ws_size; (void)n_in; (void)out_size;

  // gemm: transpose W[K,Nn] -> wtbuf[Nn,K], then strip-WMMA kernel
  auto gemm = [&](int K, const float* A, const float* A2, const float* W,
                  const float* b, float* Cc, int M, int Nn, int relu) {
    k_transpose<<<nblk((long)K * Nn), BLK, 0, stream>>>(W, wtbuf, K, Nn);
    long waves = (long)((M + 15) / 16) * (Nn / 64);
    long th = waves * 32;
    if (K == 64) {
      if (A2) k_gemm_wmma<64, true ><<<nblk(th), BLK, 0, stream>>>(A, A2, wtbuf, b, Cc, M, Nn, relu);
      else    k_gemm_wmma<64, false><<<nblk(th), BLK, 0, stream>>>(A, A2, wtbuf, b, Cc, M, Nn, relu);
    } else {
      if (A2) k_gemm_wmma<128, true ><<<nblk(th), BLK, 0, stream>>>(A, A2, wtbuf, b, Cc, M, Nn, relu);
      else    k_gemm_wmma<128, false><<<nblk(th), BLK, 0, stream>>>(A, A2, wtbuf, b, Cc, M, Nn, relu);
    }
  };
  auto bn = [&](float* H, int M, const float* g, const float* be) {
    k_zero<<<1, BLK, 0, stream>>>(stats, 2 * HID);
    k_bn_stats<<<240, HID, 0, stream>>>(H, stats, M);
    k_bn_norm_relu<<<nblk((long)M * HID), BLK, 0, stream>>>(H, stats, g, be, M);
  };

  // ---- node encoder: x[N,3] -> Linear -> BN -> ReLU -> Linear -> v0[N,64]
  k_lin_small<<<nblk((long)N * HID), BLK, 0, stream>>>(x, P(NE), P(NE + 2), hbuf, N, 3);
  bn(hbuf, N, P(NE + 5), P(NE + 4));
  gemm(128, hbuf, nullptr, P(NE + 1), P(NE + 3), v0, N, HH, 0);

  // ---- edge encoder: edge_attr[E,1] -> ... -> e0[E,64]
  k_lin_small<<<nblk((long)E * HID), BLK, 0, stream>>>(ea, P(EE), P(EE + 2), hbuf, E, 1);
  bn(hbuf, E, P(EE + 5), P(EE + 4));
  gemm(128, hbuf, nullptr, P(EE + 1), P(EE + 3), e0, E, HH, 0);

  // ---- GEN layers ----
  float* vc = v0; float* vn = v1;
  float* ecur = e0; float* en = e1;
  for (int l = 0; l < 3; ++l) {
    int nb = LN(l), eb = LE(l);
    // node branch: agg = segsum(relu(v[row]+e)+eps, col); v' = mlp(v+agg)
    k_zero<<<nblk((long)N * HH), BLK, 0, stream>>>(agg, (long)N * HH);
    k_scatter_node<<<nblk((long)E * HH), BLK, 0, stream>>>(vc, ecur, row, col, agg, E);
    gemm(64,  vc, agg, P(nb),     P(nb + 2), hbuf, N, HID, 1);
    gemm(128, hbuf, nullptr, P(nb + 1), P(nb + 3), vn, N, HH, 0);
    // edge branch on line graph: eagg = segsum(relu(e[er])+eps, ec); e' = mlp(e+eagg)
    k_zero<<<nblk((long)E * HH), BLK, 0, stream>>>(agg, (long)E * HH);
    k_scatter_edge<<<nblk(EL * HH), BLK, 0, stream>>>(ecur, er, ec, agg, EL);
    gemm(64,  ecur, agg, P(eb),     P(eb + 2), hbuf, E, HID, 1);
    gemm(128, hbuf, nullptr, P(eb + 1), P(eb + 3), en, E, HH, 0);
    float* t;
    t = vc; vc = vn; vn = t;
    t = ecur; ecur = en; en = t;
  }

  // ---- decoders (Linear -> BN -> ReLU -> Linear[128,2]) ----
  float* out_node = (float*)d_out;
  float* out_edge = (float*)d_out + (long)N * 2;

  gemm(64, vc, nullptr, P(ND), P(ND + 2), hbuf, N, HID, 0);
  bn(hbuf, N, P(ND + 5), P(ND + 4));
  k_dec_out<<<nblk((long)N * 2), BLK, 0, stream>>>(hbuf, P(ND + 1), P(ND + 3), out_node, N);

  gemm(64, ecur, nullptr, P(ED), P(ED + 2), hbuf, E, HID, 0);
  bn(hbuf, E, P(ED + 5), P(ED + 4));
  k_dec_out<<<nblk((long)E * 2), BLK, 0, stream>>>(hbuf, P(ED + 1), P(ED + 3), out_edge, E);
}